// BaseModel_75909251989855
// MI455X (gfx1250) — compile-verified
//
#include <hip/hip_runtime.h>

#define B_ 8
#define D_ 128
#define N_ 100000
#define P_ 64
#define H_ 1024
#define K_ 128
#define CHUNK 4096

typedef __attribute__((ext_vector_type(2))) float v2f;
typedef __attribute__((ext_vector_type(8))) float v8f;

// ---------------- zero scratch accumulators ----------------
__global__ void zero_kernel(float* __restrict__ sums, int* __restrict__ counts) {
  int i = blockIdx.x * blockDim.x + threadIdx.x;
  if (i < B_ * P_ * D_) sums[i] = 0.f;
  if (i < B_ * P_)      counts[i] = 0;
}

// ---------------- segment-sum pooling (bandwidth-bound core) ----------------
// One block = one (batch, n-chunk). LDS accumulator laid out acc[d*P + p] so
// the LDS bank is p%64 -> distinct labels hit distinct banks.
__global__ __launch_bounds__(256) void seg_pool_kernel(
    const float* __restrict__ feats, const int* __restrict__ labels,
    float* __restrict__ gsums, int* __restrict__ gcnt, int nchunks) {
  __shared__ float acc[D_ * P_];   // 32 KB
  __shared__ int   slab[CHUNK];    // 16 KB labels, reused across all d
  __shared__ int   scnt[P_];
  const int tid = threadIdx.x;
  const int b   = blockIdx.x / nchunks;
  const int c   = blockIdx.x % nchunks;
  const int n0  = c * CHUNK;

  for (int i = tid; i < D_ * P_; i += 256) acc[i] = 0.f;
  if (tid < P_) scnt[tid] = 0;
  __syncthreads();

  const int* lb = labels + (size_t)b * N_;
  for (int i = tid * 4; i < CHUNK; i += 256 * 4) {
    int n = n0 + i;
    if (n + 3 < N_) {
      int4 l4 = *(const int4*)(lb + n);
      slab[i] = l4.x; slab[i + 1] = l4.y; slab[i + 2] = l4.z; slab[i + 3] = l4.w;
      atomicAdd(&scnt[l4.x], 1); atomicAdd(&scnt[l4.y], 1);
      atomicAdd(&scnt[l4.z], 1); atomicAdd(&scnt[l4.w], 1);
    } else {
      for (int k = 0; k < 4; ++k) {
        int nn = n + k;
        if (nn < N_) { int l = lb[nn]; slab[i + k] = l; atomicAdd(&scnt[l], 1); }
        else slab[i + k] = 0;
      }
    }
  }
  __syncthreads();

  for (int d = 0; d < D_; ++d) {
    const float* fp = feats + ((size_t)b * D_ + d) * N_;
    float* arow = acc + d * P_;
    for (int i = tid * 4; i < CHUNK; i += 256 * 4) {
      int n = n0 + i;
      if (n + 3 < N_) {
        float4 f4 = *(const float4*)(fp + n);
        atomicAdd(&arow[slab[i]],     f4.x);
        atomicAdd(&arow[slab[i + 1]], f4.y);
        atomicAdd(&arow[slab[i + 2]], f4.z);
        atomicAdd(&arow[slab[i + 3]], f4.w);
      } else {
        for (int k = 0; k < 4; ++k) {
          int nn = n + k;
          if (nn < N_) atomicAdd(&arow[slab[i + k]], fp[nn]);
        }
      }
    }
  }
  __syncthreads();

  float* gs = gsums + (size_t)b * P_ * D_;
  for (int i = tid; i < P_ * D_; i += 256) {
    int d = i / P_, p = i % P_;
    float v = acc[i];
    if (v != 0.f) atomicAdd(&gs[p * D_ + d], v);
  }
  if (tid < P_) { int v = scnt[tid]; if (v) atomicAdd(&gcnt[b * P_ + tid], v); }
}

// ---------------- pooled = sums / max(count,1) ----------------
__global__ void pooled_kernel(const float* __restrict__ sums,
                              const int* __restrict__ counts,
                              float* __restrict__ pooled) {
  int i = blockIdx.x * blockDim.x + threadIdx.x;
  if (i >= B_ * P_ * D_) return;
  int c = counts[i / D_];
  pooled[i] = sums[i] / (float)(c ? c : 1);
}

// ---------------- GEMM1 (512x128 @ 128x1024) + bias + BN + ReLU ----------------
// One wave per 16x16 tile; fp32 WMMA 16x16x4. A/B layouts per ISA 7.12.2:
// lane l: row = l&15, K = 2*(l>>4) + v for VGPR v.
__global__ __launch_bounds__(256) void gemm1_kernel(
    const float* __restrict__ pooled, const float* __restrict__ w1,
    const float* __restrict__ b1, const float* __restrict__ gamma,
    const float* __restrict__ beta, const float* __restrict__ rmean,
    const float* __restrict__ rvar, float* __restrict__ hbuf) {
  const int wave = (blockIdx.x * blockDim.x + threadIdx.x) >> 5;
  const int lane = threadIdx.x & 31;
  const int half = lane >> 4, lm = lane & 15;
  const int NT = H_ / 16;            // 64 tiles along H
  const int m0 = (wave / NT) * 16;
  const int n0 = (wave % NT) * 16;

  v8f c = {};
  const float* ap = pooled + (size_t)(m0 + lm) * D_ + 2 * half;
  const float* bp = w1     + (size_t)(n0 + lm) * D_ + 2 * half;
  for (int kk = 0; kk < D_; kk += 4) {
    v2f a = *(const v2f*)(ap + kk);
    v2f b = *(const v2f*)(bp + kk);
    c = __builtin_amdgcn_wmma_f32_16x16x4_f32(false, a, false, b, (short)0, c,
                                              false, false);
  }
  const int j = n0 + lm;
  const float sc = gamma[j] * rsqrtf(rvar[j] + 1e-5f);
  const float bias = b1[j], mu = rmean[j], bt = beta[j];
  for (int r = 0; r < 8; ++r) {
    int m = m0 + r + 8 * half;
    float v = (c[r] + bias - mu) * sc + bt;
    hbuf[(size_t)m * H_ + j] = fmaxf(v, 0.f);
  }
}

// ---------------- GEMM2 (512x1024 @ 1024x128) ----------------
__global__ __launch_bounds__(256) void gemm2_kernel(
    const float* __restrict__ hbuf, const float* __restrict__ w2,
    float* __restrict__ out) {
  const int wave = (blockIdx.x * blockDim.x + threadIdx.x) >> 5;
  const int lane = threadIdx.x & 31;
  const int half = lane >> 4, lm = lane & 15;
  const int NT = K_ / 16;            // 8 tiles along K
  const int m0 = (wave / NT) * 16;
  const int n0 = (wave % NT) * 16;

  v8f c = {};
  const float* ap = hbuf + (size_t)(m0 + lm) * H_ + 2 * half;
  const float* bp = w2   + (size_t)(n0 + lm) * H_ + 2 * half;
  for (int kk = 0; kk < H_; kk += 4) {
    v2f a = *(const v2f*)(ap + kk);
    v2f b = *(const v2f*)(bp + kk);
    c = __builtin_amdgcn_wmma_f32_16x16x4_f32(false, a, false, b, (short)0, c,
                                              false, false);
  }
  for (int r = 0; r < 8; ++r) {
    int m = m0 + r + 8 * half;
    out[(size_t)m * K_ + n0 + lm] = c[r];
  }
}

// ---------------- row L2 normalize + counts -> int64 ----------------
__global__ __launch_bounds__(256) void norm_kernel(float* __restrict__ out,
                                                   const int* __restrict__ counts,
                                                   long long* __restrict__ cout) {
  const int wave = (blockIdx.x * blockDim.x + threadIdx.x) >> 5;
  const int lane = threadIdx.x & 31;
  if (wave >= B_ * P_) return;
  float* row = out + (size_t)wave * K_;
  float v0 = row[lane], v1 = row[lane + 32], v2 = row[lane + 64], v3 = row[lane + 96];
  float s = v0 * v0 + v1 * v1 + v2 * v2 + v3 * v3;
  for (int m = 16; m >= 1; m >>= 1) s += __shfl_xor(s, m, 32);
  float inv = 1.f / fmaxf(sqrtf(s), 1e-12f);
  row[lane] = v0 * inv; row[lane + 32] = v1 * inv;
  row[lane + 64] = v2 * inv; row[lane + 96] = v3 * inv;
  if (lane == 0) cout[wave] = (long long)counts[wave];
}

extern "C" void kernel_launch(void* const* d_in, const int* in_sizes, int n_in,
                              void* d_out, int out_size, void* d_ws, size_t ws_size,
                              hipStream_t stream) {
  const float* feats  = (const float*)d_in[0];
  const int*   labels = (const int*)d_in[1];
  const float* w1     = (const float*)d_in[2];
  const float* b1     = (const float*)d_in[3];
  const float* gamma  = (const float*)d_in[4];
  const float* beta   = (const float*)d_in[5];
  const float* rmean  = (const float*)d_in[6];
  const float* rvar   = (const float*)d_in[7];
  const float* w2     = (const float*)d_in[8];

  char* ws = (char*)d_ws;
  float* sums   = (float*)ws;                                   // 65536 f32
  int*   counts = (int*)(ws + (size_t)65536 * 4);               // 512 i32
  float* pooled = (float*)(ws + (size_t)65536 * 4 + 2048);      // 65536 f32
  float* hbuf   = (float*)(ws + (size_t)65536 * 8 + 2048);      // 512*1024 f32

  float*     outf = (float*)d_out;                // [B,P,K] f32
  long long* outc = (long long*)(outf + B_ * P_ * K_);  // [B,P] i64

  zero_kernel<<<256, 256, 0, stream>>>(sums, counts);
  const int nchunks = (N_ + CHUNK - 1) / CHUNK;   // 25
  seg_pool_kernel<<<B_ * nchunks, 256, 0, stream>>>(feats, labels, sums, counts,
                                                    nchunks);
  pooled_kernel<<<256, 256, 0, stream>>>(sums, counts, pooled);
  gemm1_kernel<<<256, 256, 0, stream>>>(pooled, w1, b1, gamma, beta, rmean, rvar,
                                        hbuf);  // 2048 tiles / 8 waves per block
  gemm2_kernel<<<32, 256, 0, stream>>>(hbuf, w2, outf);  // 256 tiles
  norm_kernel<<<64, 256, 0, stream>>>(outf, counts, outc);
}